// MotionGenerator_39591008534779
// MI455X (gfx1250) — compile-verified
//
#include <hip/hip_runtime.h>

// Gaussian RBF voting field, factorized into two rank-128 GEMMs:
//   field[c] = A (1024x128) x Bc^T (128x1024), via V_WMMA_F32_16X16X4_F32.
// Kernel 1 builds the separable exp factors; kernel 2 is the WMMA GEMM.
// Each wave computes a 16(M=x) x 32(N=y) tile (2 accumulators, shared A
// fragment); the 8 waves of a block tile a 64x64 region (4 tx x 2 ty) so
// factor reads are amortized in WGP$/L2.

#define NPTS 128
#define W 1024
#define L 1024

typedef __attribute__((ext_vector_type(2))) float v2f;
typedef __attribute__((ext_vector_type(8))) float v8f;

// ---------------------------------------------------------------------------
// Kernel 1: separable Gaussian factors.
//   Afac[x*128+n]            = exp(-(x-xc_n)^2 * k)
//   Bfac[c*W*128 + y*128+n]  = exp(-(y-yc_n)^2 * k) * v[c,n]
// ---------------------------------------------------------------------------
__global__ void rbf_factors(const float* __restrict__ v,
                            const int*   __restrict__ xc,
                            const int*   __restrict__ yc,
                            float* __restrict__ Afac,
                            float* __restrict__ Bfac)
{
    const int tid = blockIdx.x * blockDim.x + threadIdx.x;   // 0 .. W*NPTS-1
    const int n = tid & (NPTS - 1);
    const int r = tid >> 7;                                   // row 0..1023
    const float k = 1.0f / (2.0f * 32.0f * 32.0f);

    const float dx = (float)r - (float)xc[n];
    const float dy = (float)r - (float)yc[n];

    const float ey = expf(-dy * dy * k);
    Afac[tid]              = expf(-dx * dx * k);
    Bfac[tid]              = ey * v[n];          // channel 0
    Bfac[W * NPTS + tid]   = ey * v[NPTS + n];   // channel 1
}

// ---------------------------------------------------------------------------
// Kernel 2: WMMA GEMM. One wave32 -> one 16x32 output tile, K=128 in 32
// steps of V_WMMA_F32_16X16X4_F32 (2 WMMAs/step sharing the A fragment).
// Waves: 2 ch * 64 tx * 32 ty32 = 4096 -> 512 blocks of 8 waves.
// Block maps waves 2-D: wx = wave&3 (tx), wy = wave>>2 (ty) -> 64x64 region.
// ---------------------------------------------------------------------------
__global__ void __launch_bounds__(256)
rbf_wmma(const float* __restrict__ Afac,
         const float* __restrict__ Bfac,
         float* __restrict__ out)
{
    const int lane = threadIdx.x & 31;
    const int wave = threadIdx.x >> 5;

    const int b   = blockIdx.x;                  // 0..511
    const int c   = b >> 8;                      // channel
    const int rem = b & 255;
    const int bx  = rem >> 4;                    // 0..15 (x super-tile)
    const int by  = rem & 15;                    // 0..15 (y super-tile)
    const int wx  = wave & 3;                    // tx within block
    const int wy  = wave >> 2;                   // ty within block

    const int xbase = (bx * 4 + wx) << 4;        // 16-row A tile
    const int ybase = (by * 2 + wy) << 5;        // 32-col B tile

    const int lm = lane & 15;                    // M / N within 16x16 tile
    const int hi = lane >> 4;                    // K-pair select

    // fp32 WMMA fragment layout (A documented; B symmetric): lane holds the
    // float2 {K0+2*hi, K0+2*hi+1} of its row/column.
    const float* Arow  = Afac + (size_t)(xbase + lm) * NPTS + 2 * hi;
    const float* Brow0 = Bfac + (size_t)c * (W * NPTS)
                              + (size_t)(ybase + lm) * NPTS + 2 * hi;
    const float* Brow1 = Brow0 + 16 * NPTS;      // second 16-col group

    v8f acc0 = {};
    v8f acc1 = {};
#pragma unroll
    for (int kk = 0; kk < NPTS; kk += 4) {
        v2f a  = *(const v2f*)(Arow  + kk);
        v2f b0 = *(const v2f*)(Brow0 + kk);
        v2f b1 = *(const v2f*)(Brow1 + kk);
        // (neg_a, A, neg_b, B, c_mod, C, reuse_a, reuse_b)
        acc0 = __builtin_amdgcn_wmma_f32_16x16x4_f32(false, a, false, b0,
                                                     (short)0, acc0, false, false);
        acc1 = __builtin_amdgcn_wmma_f32_16x16x4_f32(false, a, false, b1,
                                                     (short)0, acc1, false, false);
    }

    // D layout: VGPR j -> row (j + 8*hi), col lm.
    float* o0 = out + ((size_t)c << 20) + (size_t)(xbase + 8 * hi) * L + (ybase + lm);
    float* o1 = o0 + 16;
#pragma unroll
    for (int j = 0; j < 8; ++j) {
        o0[(size_t)j * L] = acc0[j];
        o1[(size_t)j * L] = acc1[j];
    }
}

// ---------------------------------------------------------------------------
extern "C" void kernel_launch(void* const* d_in, const int* in_sizes, int n_in,
                              void* d_out, int out_size, void* d_ws, size_t ws_size,
                              hipStream_t stream)
{
    (void)in_sizes; (void)n_in; (void)out_size; (void)ws_size;

    const float* v  = (const float*)d_in[0];   // init_vectors [2,128]
    const int*   xc = (const int*)d_in[1];     // x_coord [128]
    const int*   yc = (const int*)d_in[2];     // y_coord [128]
    // d_in[3], d_in[4] = width/lenth scalars (fixed at 1024 here)

    float* Afac = (float*)d_ws;                // 1024*128 floats (512 KB)
    float* Bfac = Afac + W * NPTS;             // 2*1024*128 floats (1 MB)
    float* out  = (float*)d_out;               // [2,1024,1024]

    rbf_factors<<<(W * NPTS) / 256, 256, 0, stream>>>(v, xc, yc, Afac, Bfac);
    rbf_wmma<<<512, 256, 0, stream>>>(Afac, Bfac, out);
}